// ARqGPSFull_43370579755144
// MI455X (gfx1250) — compile-verified
//
#include <hip/hip_runtime.h>
#include <math.h>

// ---------------------------------------------------------------------------
// ARqGPS full model on MI455X (gfx1250).
// Log-domain reformulation: per-site complex context product ->
//   two fp32 GEMMs (log-magnitude, phase) on V_WMMA_F32_16X16X4_F32.
// ---------------------------------------------------------------------------

#define NS    64    // number of sites (L)
#define LD    2     // local dim (D)
#define MD    64    // GPS support dim (M)
#define BATCH 128   // batch size
#define KD    128   // GEMM K = L * D

typedef __attribute__((ext_vector_type(2))) float v2f;
typedef __attribute__((ext_vector_type(8))) float v8f;

// ---------------------------------------------------------------------------
// Kernel 1: params_context (complex128, [s][d][m][j]) ->
//   logRe/logIm fp32 in WMMA-friendly [s][m][K] layout, K = 2*j + d.
// Reference's mask (pad -> 1.0) becomes logP -> 0.0 here.
// ---------------------------------------------------------------------------
__global__ __launch_bounds__(256) void prep_logs_kernel(
    const double* __restrict__ pc,   // 2*NS*LD*MD*NS doubles (interleaved re,im)
    float* __restrict__ logRe,       // [NS][MD][KD]
    float* __restrict__ logIm)
{
    int t = blockIdx.x * blockDim.x + threadIdx.x;
    if (t >= NS * LD * MD * NS) return;
    int j = t & 63;
    int m = (t >> 6) & 63;
    int d = (t >> 12) & 1;
    int s = t >> 13;

    float lr = 0.f, li = 0.f;
    int cap = (s < 1) ? 1 : s;          // max(s, 1)
    if (j < cap) {
        double re = pc[2 * (size_t)t];
        double im = pc[2 * (size_t)t + 1];
        lr = (float)(0.5 * log(re * re + im * im));
        li = (float)atan2(im, re);
    }
    size_t o = ((size_t)s * MD + m) * KD + (size_t)(2 * j + d);
    logRe[o] = lr;
    logIm[o] = li;
}

// ---------------------------------------------------------------------------
// Kernel 2: one-hot A matrix, A[b][K] = (idx[b, K/2] == K%2) ? 1 : 0
// ---------------------------------------------------------------------------
__global__ __launch_bounds__(256) void prep_A_kernel(
    const int* __restrict__ idx,     // [BATCH][NS]
    float* __restrict__ A)           // [BATCH][KD]
{
    int t = blockIdx.x * blockDim.x + threadIdx.x;
    if (t >= BATCH * KD) return;
    int K = t & (KD - 1);
    int b = t >> 7;
    int j = K >> 1;
    int d = K & 1;
    A[t] = (idx[b * NS + j] == d) ? 1.0f : 0.0f;
}

// ---------------------------------------------------------------------------
// Kernel 3: per-site GEMM on WMMA + epilogue.
// grid = (8 b-tiles, 64 sites), block = 128 threads (4 waves).
// Wave w owns m-tile [16w, 16w+16). Two accumulators: Re-log and Im-log.
// Epilogue: ctx = exp(Sre) * (cos(Sim), sin(Sim)); complex-mul with
// inputs_param[s, idx[b,s], m]; LDS reduce over m; write psi[s][b].
// ---------------------------------------------------------------------------
__global__ __launch_bounds__(128) void gemm_kernel(
    const float* __restrict__ logRe,   // [NS][MD][KD]
    const float* __restrict__ logIm,
    const float* __restrict__ Amat,    // [BATCH][KD]
    const int*   __restrict__ idx,     // [BATCH][NS]
    const double* __restrict__ ip,     // inputs_param complex128 [NS][LD][MD]
    float2* __restrict__ psi)          // [NS][BATCH]
{
    const int btile = blockIdx.x;          // 0..7
    const int s     = blockIdx.y;          // 0..63
    const int wave  = threadIdx.x >> 5;    // 0..3
    const int lane  = threadIdx.x & 31;
    const int half  = lane >> 4;           // 0 or 1
    const int l16   = lane & 15;
    const int m0    = wave * 16;
    const int b0    = btile * 16;

    // B fragment rows: lane -> column n = m0+l16; K striped over halves/VGPRs.
    const float* bReRow = logRe + ((size_t)s * MD + (m0 + l16)) * KD;
    const float* bImRow = logIm + ((size_t)s * MD + (m0 + l16)) * KD;
    // A fragment rows: lane -> row M = b0+l16 (both halves), K over halves.
    const float* aRow = Amat + (size_t)(b0 + l16) * KD;

    v8f accRe = {};
    v8f accIm = {};

    for (int k0 = 0; k0 < KD; k0 += 4) {
        int off = k0 + 2 * half;           // v0 holds K=off, v1 holds K=off+1
        v2f a  = *(const v2f*)(aRow  + off);
        v2f br = *(const v2f*)(bReRow + off);
        v2f bi = *(const v2f*)(bImRow + off);
        if (k0 + 4 < KD) {                 // prefetch next K-step of B
            __builtin_prefetch(bReRow + off + 4, 0, 1);
            __builtin_prefetch(bImRow + off + 4, 0, 1);
        }
        // D = A * B + C   (16x16x4 fp32 matrix pipe)
        accRe = __builtin_amdgcn_wmma_f32_16x16x4_f32(
            false, a, false, br, (short)0, accRe, false, false);
        accIm = __builtin_amdgcn_wmma_f32_16x16x4_f32(
            false, a, false, bi, (short)0, accIm, false, false);
    }

    // ---- epilogue ----
    __shared__ float ldsRe[16][MD];
    __shared__ float ldsIm[16][MD];

    const int mcol = m0 + l16;             // this lane's m column
#pragma unroll
    for (int r = 0; r < 8; ++r) {
        int bl = r + 8 * half;             // C layout: VGPR r -> M = r / r+8
        int b  = b0 + bl;
        float mag = __expf(accRe[r]);
        float sn, cs;
        __sincosf(accIm[r], &sn, &cs);
        float cr = mag * cs;               // ctxval real
        float ci = mag * sn;               // ctxval imag

        int db = idx[b * NS + s];          // observed local state at site s
        size_t ib = (((size_t)s * LD + db) * MD + mcol) * 2;
        float ipr = (float)ip[ib];
        float ipi = (float)ip[ib + 1];

        ldsRe[bl][mcol] = cr * ipr - ci * ipi;
        ldsIm[bl][mcol] = cr * ipi + ci * ipr;
    }
    __syncthreads();

    if (threadIdx.x < 16) {                // one thread per b row: reduce over m
        float sr = 0.f, si = 0.f;
#pragma unroll 8
        for (int m = 0; m < MD; ++m) {
            sr += ldsRe[threadIdx.x][m];
            si += ldsIm[threadIdx.x][m];
        }
        psi[(size_t)s * BATCH + b0 + threadIdx.x] = make_float2(sr, si);
    }
}

// ---------------------------------------------------------------------------
// Kernel 4: deterministic sum over sites + phase wrap, write complex128.
// ---------------------------------------------------------------------------
__global__ __launch_bounds__(128) void finalize_kernel(
    const float2* __restrict__ psi,    // [NS][BATCH]
    double* __restrict__ out)          // [BATCH] complex128 interleaved
{
    int b = threadIdx.x;
    if (b >= BATCH) return;
    double sr = 0.0, si = 0.0;
    for (int s = 0; s < NS; ++s) {
        float2 v = psi[(size_t)s * BATCH + b];
        sr += (double)v.x;
        si += (double)v.y;
    }
    out[2 * b]     = sr;                       // logsumexp over n_symm=1 == Re
    out[2 * b + 1] = atan2(sin(si), cos(si));  // angle(exp(i*Im))
}

// ---------------------------------------------------------------------------
// Workspace layout (bytes):
//   [0,            2MB)  logRe   NS*MD*KD fp32
//   [2MB,          4MB)  logIm
//   [4MB,      4MB+64K)  Amat    BATCH*KD fp32
//   [4MB+64K, 4MB+128K)  psi     NS*BATCH float2
// ---------------------------------------------------------------------------
extern "C" void kernel_launch(void* const* d_in, const int* in_sizes, int n_in,
                              void* d_out, int out_size, void* d_ws, size_t ws_size,
                              hipStream_t stream) {
    const int*    idx = (const int*)d_in[0];      // (128, 64) int32
    const double* pc  = (const double*)d_in[1];   // (64,2,64,64) complex128
    const double* ip  = (const double*)d_in[2];   // (64,2,64) complex128
    double*       out = (double*)d_out;           // (128,) complex128

    char* ws = (char*)d_ws;
    float*  logRe = (float*)(ws);
    float*  logIm = (float*)(ws + (size_t)2 * 1024 * 1024);
    float*  Amat  = (float*)(ws + (size_t)4 * 1024 * 1024);
    float2* psi   = (float2*)(ws + (size_t)4 * 1024 * 1024 + 65536);

    prep_logs_kernel<<<(NS * LD * MD * NS + 255) / 256, 256, 0, stream>>>(pc, logRe, logIm);
    prep_A_kernel<<<(BATCH * KD + 255) / 256, 256, 0, stream>>>(idx, Amat);

    dim3 grid(BATCH / 16, NS);
    gemm_kernel<<<grid, 128, 0, stream>>>(logRe, logIm, Amat, idx, ip, psi);

    finalize_kernel<<<1, 128, 0, stream>>>(psi, out);
}